// ContourTransformer_76244259439048
// MI455X (gfx1250) — compile-verified
//
#include <hip/hip_runtime.h>
#include <hip/hip_bf16.h>

// ---------------- static problem config (matches reference) ----------------
#define B_    16
#define NQ_   1000
#define PS_   4
#define NP_   16
#define C_    256
#define RM_   32
#define NC_   80
#define L_    (NQ_*PS_)        // 4000 tokens per batch
#define DREG_ 264              // reg head out dim per token
#define DPAD_ 272              // DREG_ padded to 17 full 16-col tiles
#define NBIN_ (RM_+1)          // 33 bins
#define M1_   (B_*L_)          // 64000 rows  (GEMM1)
#define M2_   (B_*NQ_)         // 16000 rows  (GEMM2, post token-average)
#define PI_F  3.14159265358979323846f

typedef __attribute__((ext_vector_type(2))) float v2f;
typedef __attribute__((ext_vector_type(8))) float v8f;

__device__ __forceinline__ float sigmoidf_(float x) { return 1.0f / (1.0f + __expf(-x)); }

// ---------------------------------------------------------------------------
// Zero-pad W_reg [256,264] -> Wpad [256,272] so the GEMM1 tail N-tile needs
// no load guards (padded cols contribute 0 and are never stored).
// ---------------------------------------------------------------------------
__global__ __launch_bounds__(256) void pad_wreg_kernel(
    const float* __restrict__ W, float* __restrict__ Wp)
{
    const int idx = blockIdx.x * 256 + threadIdx.x;      // 256*272 = 69632
    if (idx >= C_ * DPAD_) return;
    const int kk = idx / DPAD_;
    const int n  = idx - kk * DPAD_;
    Wp[idx] = (n < DREG_) ? W[(size_t)kk * DREG_ + n] : 0.0f;
}

// ---------------------------------------------------------------------------
// WMMA fp32 GEMM: D[M,NCOLS] = A[M,256] @ Bm[256,BSTRIDE] + bias (+ quality)
// block = 256 threads = 8 waves. Each block stages its 16-col B slice into
// LDS (16 KB) in a pair-interleaved layout so the per-step B fragment is one
// aligned ds_load_b64. Each wave owns MTILES 16-row M tiles: one B fragment
// feeds MTILES independent V_WMMA_F32_16X16X4_F32 (no D->A/B RAW hazards).
// EXEC is all-ones at every WMMA; the only guard is on the final store.
// ---------------------------------------------------------------------------
template <int NCOLS, int BSTRIDE, int MTILES, bool EPI_SCORES>
__global__ __launch_bounds__(256) void wmma_gemm_f32(
    const float* __restrict__ A,
    const float* __restrict__ Bm,
    const float* __restrict__ bias,
    const float* __restrict__ tq,      // per-token quality (EPI_SCORES only)
    float* __restrict__ D)
{
    __shared__ float sB[C_ * 16];      // 16 KB, pair-interleaved

    const int tid  = threadIdx.x;
    const int lane = tid & 31;
    const int wave = tid >> 5;
    const int m    = lane & 15;
    const int hi   = lane >> 4;                 // 0 -> K {0,1}, 1 -> K {2,3}
    const int col0 = blockIdx.y * 16;
    const int col  = col0 + m;

    // stage B tile: element (kk,n) -> sB[((kk>>1)*16 + n)*2 + (kk&1)]
    #pragma unroll
    for (int i = 0; i < (C_ * 16) / 256; ++i) {
        const int idx = tid + i * 256;
        const int kk = idx >> 4;
        const int n  = idx & 15;
        sB[((kk >> 1) * 16 + n) * 2 + (kk & 1)] =
            Bm[(size_t)kk * BSTRIDE + col0 + n];
    }
    __syncthreads();

    const int rowBase = (blockIdx.x * 8 + wave) * (16 * MTILES);

    v8f acc[MTILES];
    #pragma unroll
    for (int jt = 0; jt < MTILES; ++jt) {
        #pragma unroll
        for (int e = 0; e < 8; ++e) acc[jt][e] = 0.0f;
    }

    const float* ap = A + (size_t)(rowBase + m) * C_ + 2 * hi;

    #pragma unroll 2
    for (int k = 0; k < C_; k += 4) {
        const v2f b = *(const v2f*)(sB + ((k >> 1) + hi) * 32 + m * 2);
        #pragma unroll
        for (int jt = 0; jt < MTILES; ++jt) {
            const v2f a = *(const v2f*)(ap + (size_t)jt * 16 * C_);
            acc[jt] = __builtin_amdgcn_wmma_f32_16x16x4_f32(
                false, a, false, b, (short)0, acc[jt], false, false);
        }
        ap += 4;
    }

    if (col < NCOLS) {
        #pragma unroll
        for (int jt = 0; jt < MTILES; ++jt) {
            #pragma unroll
            for (int r = 0; r < 8; ++r) {
                const int row = rowBase + jt * 16 + r + hi * 8;
                float v = acc[jt][r] + bias[col];
                if (EPI_SCORES) {
                    const float* t4 = tq + (size_t)row * 4;   // 4 tokens/query
                    v += 0.25f * (t4[0] + t4[1] + t4[2] + t4[3]);
                }
                D[(size_t)row * NCOLS + col] = v;
            }
        }
    }
}

// ---------------------------------------------------------------------------
// K0: average the PS=4 sub-tokens of each query (mean is linear, so the
// scores GEMM runs on averaged activations: 4x fewer FLOPs).
// ---------------------------------------------------------------------------
__global__ __launch_bounds__(256) void avg_tokens_kernel(
    const float* __restrict__ out, float* __restrict__ avg)
{
    const int c = threadIdx.x;                  // 0..255 (coalesced)
    const int q = blockIdx.x;                   // 0..999
    const int b = blockIdx.y;                   // 0..15
    const float* p = out + (size_t)(b * L_ + q * PS_) * C_ + c;
    avg[(size_t)(b * NQ_ + q) * C_ + c] =
        0.25f * (p[0] + p[C_] + p[2 * C_] + p[3 * C_]);
}

// ---------------------------------------------------------------------------
// K2: per-token epilogue. One thread per token:
//   8x { 2-pass softmax over 33 bins (pred_corners is L2-resident),
//        project-weighted integral -> contour coordinate,
//        register insertion top-8 -> LQE stat }
//   LQE MLP 72->64->1 with W_lqe1 staged in LDS (18 KB), h[64] in VGPRs.
// ---------------------------------------------------------------------------
__global__ __launch_bounds__(128) void epilogue_kernel(
    const float* __restrict__ pc,        // pred_corners [M1_, 264]
    const float* __restrict__ refu,      // [B,NQ,4]
    const float* __restrict__ W1,        // [72,64]
    const float* __restrict__ b1,        // [64]
    const float* __restrict__ W2,        // [64]
    const float* __restrict__ b2,        // [1]
    float* __restrict__ coords,          // [B,NQ,NP,2]  (d_out part 1)
    float* __restrict__ tq)              // [M1_] per-token quality
{
    __shared__ float sW1[72 * 64];
    __shared__ float sb1[64];
    __shared__ float sW2[64];
    __shared__ float sproj[NBIN_];

    for (int i = threadIdx.x; i < 72 * 64; i += blockDim.x) sW1[i] = W1[i];
    if (threadIdx.x < 64) {
        sb1[threadIdx.x] = b1[threadIdx.x];
        sW2[threadIdx.x] = W2[threadIdx.x];
    }
    if (threadIdx.x < NBIN_) {
        // D-FINE weighting function, up=0.5, reg_scale=4 -> ub1=2, ub2=4,
        // step = 3^(1/15)
        const int j = threadIdx.x;
        const float step = __powf(3.0f, 1.0f / 15.0f);
        float v;
        if (j == 0)       v = -4.0f;
        else if (j == 32) v =  4.0f;
        else if (j == 16) v =  0.0f;
        else if (j < 16)  v = 1.0f - __powf(step, (float)(16 - j));
        else              v = __powf(step, (float)(j - 16)) - 1.0f;
        sproj[j] = v;
    }
    __syncthreads();

    const int t = blockIdx.x * blockDim.x + threadIdx.x;   // token id, exact fit
    const int b = t / L_;
    const int l = t - b * L_;
    const int q = l >> 2;
    const int s = l & 3;

    const float* rp = pc + (size_t)t * DREG_;
    const float* rf = refu + (size_t)(b * NQ_ + q) * 4;
    const float cx = sigmoidf_(rf[0]);
    const float cy = sigmoidf_(rf[1]);
    const float hw = 0.5f * sigmoidf_(rf[2]);
    const float hh = 0.5f * sigmoidf_(rf[3]);

    float h[64];
    #pragma unroll
    for (int o = 0; o < 64; ++o) h[o] = sb1[o];

    #pragma unroll 1
    for (int g = 0; g < 8; ++g) {
        const float* gp = rp + g * NBIN_;
        // pass 1: row max
        float mx = -INFINITY;
        #pragma unroll 1
        for (int j = 0; j < NBIN_; ++j) mx = fmaxf(mx, gp[j]);
        // pass 2: exp, sum, weighted sum, unnormalized top-8 (order-preserving)
        float top[8];
        #pragma unroll
        for (int i = 0; i < 8; ++i) top[i] = -1.0f;   // exp >= 0
        float sum = 0.0f, dist = 0.0f;
        #pragma unroll 1
        for (int j = 0; j < NBIN_; ++j) {
            const float e = __expf(gp[j] - mx);
            sum  += e;
            dist += e * sproj[j];
            if (e > top[7]) {
                float v = e;
                #pragma unroll
                for (int i = 0; i < 8; ++i) {
                    if (v > top[i]) { const float tmp = top[i]; top[i] = v; v = tmp; }
                }
            }
        }
        const float inv = 1.0f / sum;
        dist *= inv;
        float tmean = 0.0f;
        #pragma unroll
        for (int i = 0; i < 8; ++i) { top[i] *= inv; tmean += top[i]; }
        tmean *= 0.125f;

        // accumulate 9 stat values into the 64 hidden units
        const int gi = g * 9;
        #pragma unroll 1
        for (int i = 0; i < 9; ++i) {
            const float sv = (i < 8) ? top[i] : tmean;
            const float* wrow = &sW1[(gi + i) * 64];
            #pragma unroll
            for (int o = 0; o < 64; ++o) h[o] = __fmaf_rn(sv, wrow[o], h[o]);
        }

        // contour coordinate for this (s,g): p = (s*8+g)>>1, xy = g&1
        const int pg = s * 8 + g;
        const int p  = pg >> 1;
        const int xy = pg & 1;
        const float ang = PI_F * (float)p * 0.125f;     // 2*pi*p/16
        const float base = xy ? (cy + hh * __cosf(ang))
                              : (cx + hw * __sinf(ang));
        coords[((size_t)(b * NQ_ + q) * NP_ + p) * 2 + xy] = base + dist * 0.25f;
    }

    float qual = b2[0];
    #pragma unroll
    for (int o = 0; o < 64; ++o) qual += fmaxf(h[o], 0.0f) * sW2[o];
    tq[t] = qual;
}

// ---------------------------------------------------------------------------
extern "C" void kernel_launch(void* const* d_in, const int* in_sizes, int n_in,
                              void* d_out, int out_size, void* d_ws, size_t ws_size,
                              hipStream_t stream)
{
    const float* output  = (const float*)d_in[0];
    const float* refu    = (const float*)d_in[1];
    const float* W_reg   = (const float*)d_in[2];
    const float* b_reg   = (const float*)d_in[3];
    const float* W_score = (const float*)d_in[4];
    const float* b_score = (const float*)d_in[5];
    const float* W_lqe1  = (const float*)d_in[6];
    const float* b_lqe1  = (const float*)d_in[7];
    const float* W_lqe2  = (const float*)d_in[8];
    const float* b_lqe2  = (const float*)d_in[9];

    float* coords = (float*)d_out;                                  // [16,1000,16,2]
    float* scores = (float*)d_out + (size_t)B_ * NQ_ * NP_ * 2;     // [16,1000,80]

    // workspace: pred_corners (67.6MB) | avg_out (16.4MB) | tq (0.26MB) | Wpad (0.28MB)
    float* pred_corners = (float*)d_ws;
    float* avg_out      = pred_corners + (size_t)M1_ * DREG_;
    float* tq           = avg_out + (size_t)M2_ * C_;
    float* Wpad         = tq + (size_t)M1_;

    // P0: zero-pad W_reg -> [256,272]
    pad_wreg_kernel<<<(C_ * DPAD_ + 255) / 256, 256, 0, stream>>>(W_reg, Wpad);
    // K0: token-average activations for the scores GEMM
    avg_tokens_kernel<<<dim3(NQ_, B_), C_, 0, stream>>>(output, avg_out);

    // K1: pred_corners = output @ W_reg + b_reg   (fp32 WMMA, 4 M-tiles/wave)
    {
        dim3 grid(M1_ / 512, DPAD_ / 16);          // (125, 17)
        wmma_gemm_f32<DREG_, DPAD_, 4, false><<<grid, 256, 0, stream>>>(
            output, Wpad, b_reg, nullptr, pred_corners);
    }
    // K2: softmax / integral / top-k / LQE MLP / coords
    epilogue_kernel<<<M1_ / 128, 128, 0, stream>>>(
        pred_corners, refu, W_lqe1, b_lqe1, W_lqe2, b_lqe2, coords, tq);
    // K3: scores = avg_out @ W_score + b_score + quality   (fp32 WMMA)
    {
        dim3 grid(M2_ / 128, NC_ / 16);            // (125, 5)
        wmma_gemm_f32<NC_, NC_, 1, true><<<grid, 256, 0, stream>>>(
            avg_out, W_score, b_score, tq, scores);
    }
}